// Top1Gate_24653112279120
// MI455X (gfx1250) — compile-verified
//
#include <hip/hip_runtime.h>
#include <stdint.h>

typedef __attribute__((ext_vector_type(2))) float v2f;
typedef __attribute__((ext_vector_type(8))) float v8f;

#define S_TOK   262144
#define DMODEL  1024
#define NEXP    64
#define TB      128            // tokens per workgroup (8 waves x 16 tokens)
#define NBLK    (S_TOK / TB)   // 2048
#define KC      64             // K-chunk columns staged in LDS per iteration
#define NCHUNK  (DMODEL / KC)  // 16
#define XS_STRIDE 68           // pad keeps 16B alignment for b128 async stores
#define WG_STRIDE 68
#define LG_STRIDE 65           // odd stride -> bank rotation for row reads

#define XS_FLOATS (TB * XS_STRIDE)    // 8704 floats per buffer
#define WG_FLOATS (NEXP * WG_STRIDE)  // 4352 floats per buffer
// dynamic LDS: xs[2] | wgs[2] | maxv | invd | eidx   (lg reuses xs[0..])
#define SMEM_FLOATS (2 * XS_FLOATS + 2 * WG_FLOATS + 3 * TB)
#define SMEM_BYTES  (SMEM_FLOATS * 4)

// CDNA5 async global->LDS copy (ASYNCcnt-tracked, no VGPR data staging).
// GV mode: %0 = LDS byte offset (generic-ptr low 32 bits), %1 = 64-bit vaddr.
__device__ __forceinline__ void async_copy_b128(void* lds_ptr, const float* gptr) {
  uint32_t lds_off = (uint32_t)(uintptr_t)lds_ptr;   // LDS aperture: addr[31:0]
  asm volatile("global_load_async_to_lds_b128 %0, %1, off"
               :: "v"(lds_off), "v"(gptr) : "memory");
}
__device__ __forceinline__ void wait_async_12() {
  asm volatile("s_wait_asynccnt 0xc" ::: "memory");
}
__device__ __forceinline__ void wait_async_0() {
  asm volatile("s_wait_asynccnt 0x0" ::: "memory");
}

// ---------------------------------------------------------------------------
// Kernel A: WMMA GEMM (logits) + per-token gating + per-block stats
// ---------------------------------------------------------------------------
__global__ __launch_bounds__(256) void gate_gemm_kernel(
    const float* __restrict__ x, const float* __restrict__ wg,
    float* __restrict__ out, int* __restrict__ cnt, float* __restrict__ mep)
{
  extern __shared__ float smf[];
  // buffer addresses derived arithmetically (no LDS-pointer arrays!)
  float* lg   = smf;                                  // reuse xs region post-GEMM
  float* maxv = smf + 2 * XS_FLOATS + 2 * WG_FLOATS;
  float* invd = maxv + TB;
  int*   eidx = (int*)(invd + TB);

  const int tid  = threadIdx.x;
  const int blk  = blockIdx.x;
  const int tok0 = blk * TB;
  const int wave = tid >> 5;
  const int lane = tid & 31;
  const int hi   = lane >> 4;   // lane-half: K {0,1} vs K {2,3}
  const int l15  = lane & 15;

  v8f acc[4];
  #pragma unroll
  for (int t = 0; t < 4; ++t)
    #pragma unroll
    for (int j = 0; j < 8; ++j) acc[t][j] = 0.0f;

  // ---- stage one K-chunk (12 async b128 ops per thread, all in flight) ----
  auto stage = [&](int p, int cb) {
    float* xs  = smf + p * XS_FLOATS;
    float* wgs = smf + 2 * XS_FLOATS + p * WG_FLOATS;
    #pragma unroll
    for (int i = 0; i < 8; ++i) {                // x tile: 128 x 64
      int g = tid + i * 256;
      int row = g >> 4, c4 = (g & 15) << 2;
      async_copy_b128(&xs[row * XS_STRIDE + c4],
                      x + (size_t)(tok0 + row) * DMODEL + cb + c4);
    }
    #pragma unroll
    for (int i = 0; i < 4; ++i) {                // wg chunk: 64 x 64 (L2-hot)
      int g = tid + i * 256;
      int row = g >> 4, c4 = (g & 15) << 2;
      async_copy_b128(&wgs[row * WG_STRIDE + c4],
                      wg + (size_t)row * DMODEL + cb + c4);
    }
  };

  stage(0, 0);                                   // prologue: chunk 0 -> buf 0
  for (int ci = 0; ci < NCHUNK; ++ci) {
    const int p = ci & 1;
    if (ci + 1 < NCHUNK) {                       // prefetch next chunk
      stage(1 - p, (ci + 1) * KC);
      wait_async_12();                           // chunk ci landed; next in flight
    } else {
      wait_async_0();
    }
    __syncthreads();

    const float* xsb = smf + p * XS_FLOATS;
    const float* wgb = smf + 2 * XS_FLOATS + p * WG_FLOATS;
    const float* xrow = xsb + (wave * 16 + l15) * XS_STRIDE;
    const float* w0   = wgb + (l15     ) * WG_STRIDE;
    const float* w1   = wgb + (l15 + 16) * WG_STRIDE;
    const float* w2   = wgb + (l15 + 32) * WG_STRIDE;
    const float* w3   = wgb + (l15 + 48) * WG_STRIDE;
    #pragma unroll
    for (int kk = 0; kk < 16; ++kk) {
      // A 16x4 / B 4x16 f32 layouts: VGPR0 = K{0|2}, VGPR1 = K{1|3} by lane-half
      int c = kk * 4 + 2 * hi;
      v2f a;  a.x  = xrow[c]; a.y  = xrow[c + 1];
      v2f b0; b0.x = w0[c];   b0.y = w0[c + 1];
      v2f b1; b1.x = w1[c];   b1.y = w1[c + 1];
      v2f b2; b2.x = w2[c];   b2.y = w2[c + 1];
      v2f b3; b3.x = w3[c];   b3.y = w3[c + 1];
      acc[0] = __builtin_amdgcn_wmma_f32_16x16x4_f32(false, a, false, b0, (short)0, acc[0], false, false);
      acc[1] = __builtin_amdgcn_wmma_f32_16x16x4_f32(false, a, false, b1, (short)0, acc[1], false, false);
      acc[2] = __builtin_amdgcn_wmma_f32_16x16x4_f32(false, a, false, b2, (short)0, acc[2], false, false);
      acc[3] = __builtin_amdgcn_wmma_f32_16x16x4_f32(false, a, false, b3, (short)0, acc[3], false, false);
    }
    __syncthreads();                             // buf p consumed; refill at ci+1
  }

  // ---- spill logits to LDS row-major [token][expert] ----
  #pragma unroll
  for (int t = 0; t < 4; ++t)
    #pragma unroll
    for (int j = 0; j < 8; ++j) {
      int row = wave * 16 + j + 8 * hi;          // C layout: VGPR j -> M=j / j+8
      lg[row * LG_STRIDE + t * 16 + l15] = acc[t][j];
    }
  __syncthreads();

  // ---- per-token argmax + softmax gate ----
  if (tid < TB) {
    const float* row = &lg[tid * LG_STRIDE];
    float m = row[0]; int am = 0;
    #pragma unroll 8
    for (int e = 1; e < NEXP; ++e) { float v = row[e]; if (v > m) { m = v; am = e; } }
    float s = 0.0f;
    #pragma unroll 8
    for (int e = 0; e < NEXP; ++e) s += __expf(row[e] - m);
    float inv = 1.0f / s;                        // gate of argmax = exp(0)/s
    eidx[tid] = am; maxv[tid] = m; invd[tid] = inv;
    out[1 + tok0 + tid]         = inv;           // gates1_s
    out[1 + S_TOK + tok0 + tid] = (float)am;     // indices1_s
  }
  __syncthreads();

  if (tid < TB) {
    // deterministic within-block rank (tokens before me with same expert)
    int my = eidx[tid]; int r = 0;
    for (int t2 = 0; t2 < tid; ++t2) r += (eidx[t2] == my);
    out[1 + 2 * S_TOK + tok0 + tid] = (float)r;  // staged; finalized in kernel C
  } else if (tid < TB + NEXP) {
    // per-expert block histogram + fixed-order partial me sum (deterministic)
    int e = tid - TB;
    int c = 0; float me = 0.0f;
    for (int t2 = 0; t2 < TB; ++t2) {
      c  += (eidx[t2] == e);
      me += __expf(lg[t2 * LG_STRIDE + e] - maxv[t2]) * invd[t2];
    }
    cnt[blk * NEXP + e] = c;
    mep[blk * NEXP + e] = me;
  }
}

// ---------------------------------------------------------------------------
// Kernel B: exclusive scan of block counts per expert (in place), ce, loss
// ---------------------------------------------------------------------------
__global__ __launch_bounds__(64) void gate_scan_kernel(
    int* __restrict__ cnt, const float* __restrict__ mep, float* __restrict__ out)
{
  __shared__ float part[NEXP];
  int e = threadIdx.x;
  int run = 0; float me = 0.0f;
  for (int b = 0; b < NBLK; ++b) {
    int c = cnt[b * NEXP + e];
    cnt[b * NEXP + e] = run;     // exclusive prefix for this block
    run += c;
    me  += mep[b * NEXP + e];    // fixed-order -> deterministic
  }
  part[e] = me * (float)run;     // me[e] * ce[e]
  __syncthreads();
  if (e == 0) {
    float L = 0.0f;
    for (int i = 0; i < NEXP; ++i) L += part[i];
    out[0] = L * ((float)NEXP / ((float)S_TOK * (float)S_TOK));
  }
}

// ---------------------------------------------------------------------------
// Kernel C: locations = within-block rank + per-expert block prefix
// ---------------------------------------------------------------------------
__global__ __launch_bounds__(256) void gate_loc_kernel(
    const int* __restrict__ cnt, float* __restrict__ out)
{
  int t = blockIdx.x * 256 + threadIdx.x;
  int e = (int)out[1 + S_TOK + t];
  int b = t / TB;
  int loc = (int)out[1 + 2 * S_TOK + t] + cnt[b * NEXP + e];
  out[1 + 2 * S_TOK + t] = (float)loc;
}

// ---------------------------------------------------------------------------
extern "C" void kernel_launch(void* const* d_in, const int* in_sizes, int n_in,
                              void* d_out, int out_size, void* d_ws, size_t ws_size,
                              hipStream_t stream)
{
  const float* x  = (const float*)d_in[0];   // [S, D] fp32
  const float* wg = (const float*)d_in[1];   // [E, D] fp32
  float* out = (float*)d_out;                // [1 + 3*S] packed outputs

  int*   cnt = (int*)d_ws;                         // [NBLK][NEXP] counts->prefix
  float* mep = (float*)d_ws + (size_t)NBLK * NEXP; // [NBLK][NEXP] me partials

  gate_gemm_kernel<<<NBLK, 256, SMEM_BYTES, stream>>>(x, wg, out, cnt, mep);
  gate_scan_kernel<<<1, 64, 0, stream>>>(cnt, mep, out);
  gate_loc_kernel<<<S_TOK / 256, 256, 0, stream>>>(cnt, out);
}